// Net_42322607735202
// MI455X (gfx1250) — compile-verified
//
#include <hip/hip_runtime.h>
#include <hip/hip_bf16.h>

// ---------------------------------------------------------------------------
// GCN forward: N=12000 nodes, E=384000 edges, F_IN=500, HID=16, C=7
// Outputs: logp [N,7] then x_product = z @ z^T [N,N], concatenated in d_out.
// Dominant cost: 576 MB store of z@z^T -> bandwidth bound; use f32 WMMA
// (V_WMMA_F32_16X16X4_F32, K padded 7->8) for the decode GEMM.
// ---------------------------------------------------------------------------

#define GCN_N     12000
#define GCN_E     384000
#define GCN_FIN   500
#define GCN_HID   16
#define GCN_C     7
#define GCN_CP    8            // padded class dim (col 7 == 0)

// workspace layout (float offsets)
#define WS_DIS    0                    // N
#define WS_H      12032                // N*16
#define WS_HAGG   204032               // N*16
#define WS_T      396032               // N*8
#define WS_ZAGG   492032               // N*8
#define WS_Z      588032               // N*8

typedef __attribute__((ext_vector_type(2))) float v2f;
typedef __attribute__((ext_vector_type(8))) float v8f;

// ---------------- 1: init deg=1 (self loop), zero accumulators -------------
__global__ void init_kernel(float* __restrict__ deg, float* __restrict__ hagg,
                            float* __restrict__ zagg) {
    int i = blockIdx.x * blockDim.x + threadIdx.x;     // over N*16
    if (i < GCN_N * GCN_HID) hagg[i] = 0.0f;
    if (i < GCN_N * GCN_CP)  zagg[i] = 0.0f;
    if (i < GCN_N)           deg[i]  = 1.0f;           // self-loop
}

// ---------------- 2: in-degree via atomics ---------------------------------
__global__ void deg_kernel(const int* __restrict__ ei, float* __restrict__ deg) {
    int e = blockIdx.x * blockDim.x + threadIdx.x;
    if (e < GCN_E) atomicAdd(&deg[ei[GCN_E + e]], 1.0f);
}

__global__ void rsqrt_kernel(float* __restrict__ deg) {
    int i = blockIdx.x * blockDim.x + threadIdx.x;
    if (i < GCN_N) deg[i] = rsqrtf(deg[i]);            // deg >= 1 always
}

// ---------------- 3: h = x @ W1  (12000x500 @ 500x16) ----------------------
// block = 256 threads = 16 nodes x 16 cols; W1 staged in LDS (32 KB)
__global__ void gemm1_kernel(const float* __restrict__ x,
                             const float* __restrict__ W1,
                             float* __restrict__ h) {
    __shared__ float w[GCN_FIN * GCN_HID];
    for (int k = threadIdx.x; k < GCN_FIN * GCN_HID; k += 256) w[k] = W1[k];
    __syncthreads();
    int node = blockIdx.x * 16 + (threadIdx.x >> 4);
    int col  = threadIdx.x & 15;
    if (node >= GCN_N) return;
    const float* xr = x + (long long)node * GCN_FIN;
    float acc = 0.0f;
    #pragma unroll 4
    for (int k = 0; k < GCN_FIN; ++k) acc = fmaf(xr[k], w[k * GCN_HID + col], acc);
    h[node * GCN_HID + col] = acc;
}

// ---------------- 4: layer-1 scatter: hagg[dst] += h[src]*norm -------------
__global__ void agg1_kernel(const int* __restrict__ ei,
                            const float* __restrict__ dis,
                            const float* __restrict__ h,
                            float* __restrict__ hagg) {
    long long i = (long long)blockIdx.x * blockDim.x + threadIdx.x;
    long long tot = (long long)(GCN_E + GCN_N) * GCN_HID;
    if (i >= tot) return;
    int e = (int)(i >> 4);
    int f = (int)(i & 15);
    int s, d;
    if (e < GCN_E) { s = ei[e]; d = ei[GCN_E + e]; }
    else           { s = d = e - GCN_E; }
    float nrm = dis[s] * dis[d];
    atomicAdd(&hagg[d * GCN_HID + f], h[s * GCN_HID + f] * nrm);
}

// ---------------- 5: t = relu(hagg + b1) @ W2  (N x 8, col 7 = 0) ----------
__global__ void gemm2_kernel(const float* __restrict__ hagg,
                             const float* __restrict__ b1,
                             const float* __restrict__ W2,
                             float* __restrict__ t) {
    int i = blockIdx.x * blockDim.x + threadIdx.x;     // over N*8
    if (i >= GCN_N * GCN_CP) return;
    int n = i >> 3, c = i & 7;
    if (c == GCN_C) { t[i] = 0.0f; return; }
    const float* hr = hagg + n * GCN_HID;
    float acc = 0.0f;
    #pragma unroll
    for (int j = 0; j < GCN_HID; ++j) {
        float hv = fmaxf(hr[j] + b1[j], 0.0f);
        acc = fmaf(hv, W2[j * GCN_C + c], acc);
    }
    t[i] = acc;
}

// ---------------- 6: layer-2 scatter ---------------------------------------
__global__ void agg2_kernel(const int* __restrict__ ei,
                            const float* __restrict__ dis,
                            const float* __restrict__ t,
                            float* __restrict__ zagg) {
    long long i = (long long)blockIdx.x * blockDim.x + threadIdx.x;
    long long tot = (long long)(GCN_E + GCN_N) * GCN_CP;
    if (i >= tot) return;
    int e = (int)(i >> 3);
    int f = (int)(i & 7);
    if (f >= GCN_C) return;                            // padded col stays 0
    int s, d;
    if (e < GCN_E) { s = ei[e]; d = ei[GCN_E + e]; }
    else           { s = d = e - GCN_E; }
    float nrm = dis[s] * dis[d];
    atomicAdd(&zagg[d * GCN_CP + f], t[s * GCN_CP + f] * nrm);
}

// ---------------- 7: z = zagg + b2 (padded), 8-wide ------------------------
__global__ void zbias_kernel(const float* __restrict__ zagg,
                             const float* __restrict__ b2,
                             float* __restrict__ z) {
    int i = blockIdx.x * blockDim.x + threadIdx.x;
    if (i >= GCN_N * GCN_CP) return;
    int c = i & 7;
    z[i] = (c < GCN_C) ? (zagg[i] + b2[c]) : 0.0f;
}

// ---------------- 8: log_softmax over 7 classes ----------------------------
__global__ void logsoftmax_kernel(const float* __restrict__ z,
                                  float* __restrict__ logp) {
    int n = blockIdx.x * blockDim.x + threadIdx.x;
    if (n >= GCN_N) return;
    const float* zr = z + n * GCN_CP;
    float m = zr[0];
    #pragma unroll
    for (int c = 1; c < GCN_C; ++c) m = fmaxf(m, zr[c]);
    float s = 0.0f;
    #pragma unroll
    for (int c = 0; c < GCN_C; ++c) s += expf(zr[c] - m);
    float ls = logf(s);
    #pragma unroll
    for (int c = 0; c < GCN_C; ++c) logp[n * GCN_C + c] = zr[c] - m - ls;
}

// ---------------- 9: x_product = z @ z^T via V_WMMA_F32_16X16X4_F32 --------
// Each wave computes a 16x32 output strip (two 16x16 tiles, A reused).
// A(16x4 f32) per-lane layout: lanes 0-15 hold K=0(v0),1(v1); lanes 16-31
// hold K=2,3.  B(4x16) is the lane-transposed analog, and since B = z^T the
// B load pattern is identical to A's with the column-tile base.
__global__ void zzt_wmma_kernel(const float* __restrict__ z,
                                float* __restrict__ out) {
    const int lane = threadIdx.x & 31;
    const int wave = threadIdx.x >> 5;
    const int wid  = blockIdx.x * 8 + wave;            // 8 waves / block
    const int NGRP = GCN_N / 32;                       // 375 column groups
    int m_tile = wid / NGRP;
    int n_grp  = wid - m_tile * NGRP;
    if (m_tile >= GCN_N / 16) return;

    const int l15  = lane & 15;
    const int hi   = lane >> 4;                        // 0: K=0,1  1: K=2,3
    const int ksel = hi << 1;

    // A operands for k-steps 0..3 and 4..7
    const float* za = z + (m_tile * 16 + l15) * GCN_CP + ksel;
    v2f a0 = *(const v2f*)(za);
    v2f a1 = *(const v2f*)(za + 4);

    #pragma unroll
    for (int t = 0; t < 2; ++t) {
        int col0 = n_grp * 32 + t * 16;
        const float* zb = z + (col0 + l15) * GCN_CP + ksel;
        v2f b0 = *(const v2f*)(zb);
        v2f b1v = *(const v2f*)(zb + 4);
        v8f c = {};
        c = __builtin_amdgcn_wmma_f32_16x16x4_f32(false, a0, false, b0,
                                                  (short)0, c, false, false);
        c = __builtin_amdgcn_wmma_f32_16x16x4_f32(false, a1, false, b1v,
                                                  (short)0, c, false, false);
        // C/D layout: VGPR r, lanes 0-15 -> row r; lanes 16-31 -> row r+8
        long long row0 = (long long)(m_tile * 16 + (hi << 3));
        long long base = row0 * GCN_N + col0 + l15;
        #pragma unroll
        for (int r = 0; r < 8; ++r)
            out[base + (long long)r * GCN_N] = c[r];
    }
}

// ---------------------------------------------------------------------------
extern "C" void kernel_launch(void* const* d_in, const int* in_sizes, int n_in,
                              void* d_out, int out_size, void* d_ws, size_t ws_size,
                              hipStream_t stream) {
    const float* x  = (const float*)d_in[0];
    const int*   ei = (const int*)d_in[1];
    const float* W1 = (const float*)d_in[2];
    const float* b1 = (const float*)d_in[3];
    const float* W2 = (const float*)d_in[4];
    const float* b2 = (const float*)d_in[5];

    float* ws   = (float*)d_ws;
    float* dis  = ws + WS_DIS;
    float* h    = ws + WS_H;
    float* hagg = ws + WS_HAGG;
    float* t    = ws + WS_T;
    float* zagg = ws + WS_ZAGG;
    float* z    = ws + WS_Z;

    float* logp = (float*)d_out;                       // [N,7]
    float* xprod = (float*)d_out + GCN_N * GCN_C;      // [N,N]

    const int B = 256;

    init_kernel<<<(GCN_N * GCN_HID + B - 1) / B, B, 0, stream>>>(dis, hagg, zagg);
    deg_kernel<<<(GCN_E + B - 1) / B, B, 0, stream>>>(ei, dis);
    rsqrt_kernel<<<(GCN_N + B - 1) / B, B, 0, stream>>>(dis);
    gemm1_kernel<<<GCN_N / 16, B, 0, stream>>>(x, W1, h);

    long long a1t = (long long)(GCN_E + GCN_N) * GCN_HID;
    agg1_kernel<<<(int)((a1t + B - 1) / B), B, 0, stream>>>(ei, dis, h, hagg);

    gemm2_kernel<<<(GCN_N * GCN_CP + B - 1) / B, B, 0, stream>>>(hagg, b1, W2, t);

    long long a2t = (long long)(GCN_E + GCN_N) * GCN_CP;
    agg2_kernel<<<(int)((a2t + B - 1) / B), B, 0, stream>>>(ei, dis, t, zagg);

    zbias_kernel<<<(GCN_N * GCN_CP + B - 1) / B, B, 0, stream>>>(zagg, b2, z);
    logsoftmax_kernel<<<(GCN_N + B - 1) / B, B, 0, stream>>>(z, logp);

    int waves = (GCN_N / 16) * (GCN_N / 32);           // 750 * 375
    zzt_wmma_kernel<<<(waves + 7) / 8, B, 0, stream>>>(z, xprod);
}